// ALSHConvNet_81896436400437
// MI455X (gfx1250) — compile-verified
//
#include <hip/hip_runtime.h>
#include <hip/hip_bf16.h>

// ---------------------------------------------------------------------------
// ALSHConvNet for MI455X (gfx1250).  Memory-bound pipeline: h1 (165MB)
// materialized once (fits in 192MB L2); conv2+mask+relu+linear fused so h2
// never touches HBM.  Linear 1000->10 contraction uses V_WMMA_F32_16X16X4_F32
// (codegen-confirmed).  conv2 operands are LDS-tiled (40-col chunks) so the
// 45-tap / 5-wave reuse comes out of LDS banks instead of L1/L2, with
// global_prefetch_b8 software prefetch of the next chunk.
// ---------------------------------------------------------------------------

typedef float  v2f  __attribute__((ext_vector_type(2)));
typedef float  v8f  __attribute__((ext_vector_type(8)));

// Problem constants
#define BATCH   32
#define C_IN    3
#define H_IN    260
#define W_IN    1004
#define COUT1   5
#define H1R     258      // conv1 out rows
#define W1C     1002     // conv1 out cols
#define COUT2   5
#define H2R     256      // conv2 out rows
#define W2C     1000     // conv2 out cols (== linear K)
#define NLIN    10       // linear out features

#define WCHUNK  40                       // conv2 K-chunk (divides 1000, mult of 4)
#define NCHUNK  (W2C / WCHUNK)           // 25
#define TROWS   18                       // 16 out rows + 2 halo
#define TCOLS   (WCHUNK + 2)             // 42
#define TELEMS  (COUT1 * TROWS * TCOLS)  // 3780

#define H1_ELEMS ((size_t)BATCH * COUT1 * H1R * W1C)   // 41,362,560 floats

// ===========================================================================
// K1/K4: 9-shift windowed sums for ALSH patch means.
// One block per (b,c); block = 256 threads; no atomics (block owns its 9 sums).
// out[(b*C+c)*9 + i*3+j] = sum over window (i..i+Ho-1, j..j+Wo-1) of src[b,c].
// ===========================================================================
__global__ __launch_bounds__(256)
void patchsum_kernel(const float* __restrict__ src, int C, int H, int W,
                     int Ho, int Wo, float* __restrict__ out)
{
    const int bc  = blockIdx.x;                   // b*C + c
    const int tid = threadIdx.x;
    const float* p = src + (size_t)bc * H * W;
    const int N = H * W;

    float s[9];
#pragma unroll
    for (int k = 0; k < 9; ++k) s[k] = 0.f;

    for (int i = tid; i < N; i += 256) {
        const int r = i / W, w = i % W;
        const float v = p[i];
#pragma unroll
        for (int ii = 0; ii < 3; ++ii)
#pragma unroll
            for (int jj = 0; jj < 3; ++jj) {
                const bool in =
                    ((unsigned)(r - ii) < (unsigned)Ho) &&
                    ((unsigned)(w - jj) < (unsigned)Wo);
                s[ii * 3 + jj] += in ? v : 0.f;
            }
    }

    __shared__ float red[9][256];
#pragma unroll
    for (int k = 0; k < 9; ++k) red[k][tid] = s[k];
    __syncthreads();
    for (int off = 128; off > 0; off >>= 1) {
        if (tid < off) {
#pragma unroll
            for (int k = 0; k < 9; ++k) red[k][tid] += red[k][tid + off];
        }
        __syncthreads();
    }
    if (tid == 0) {
#pragma unroll
        for (int k = 0; k < 9; ++k) out[bc * 9 + k] = red[k][0];
    }
}

// ===========================================================================
// K2/K5: hash kernel.  Lanes 0..31: query hash qh[b] from patch means.
// Lane 0: filter hashes fh[f] (Cout<=5, d<=45 — trivial work).
// R = 0.1, TABLE_SIZE = 2, U = 0.99, M = 5 asymmetric dims (Q appends 0.5).
// ===========================================================================
__global__ __launch_bounds__(64)
void hash_kernel(const float* __restrict__ psums, int C, float invCount,
                 const float* __restrict__ Wt, int Cout,
                 const float* __restrict__ av, const float* __restrict__ bsc,
                 int* __restrict__ qh, int* __restrict__ fh)
{
    const int d  = C * 9;
    const float bb = bsc[0];
    const int t  = threadIdx.x;

    if (t < BATCH) {
        const float* q = psums + (size_t)t * d;
        float s2 = 0.f;
        for (int i = 0; i < d; ++i) { const float v = q[i] * invCount; s2 += v * v; }
        const float nrm = fmaxf(sqrtf(s2), 1e-12f);
        float dot = 0.f;
        for (int i = 0; i < d; ++i) dot += (q[i] * invCount / nrm) * av[i];
#pragma unroll
        for (int m = 0; m < 5; ++m) dot += 0.5f * av[d + m];
        const int hi = (int)floorf((dot + bb) * 10.0f);   // /R, R=0.1
        qh[t] = hi & 1;                                    // mod 2, non-negative
    }
    if (t == 0) {
        float norms[8];
        float mx = 0.f;
        for (int f = 0; f < Cout; ++f) {
            float s = 0.f;
            for (int i = 0; i < d; ++i) { const float v = Wt[f * d + i]; s += v * v; }
            norms[f] = sqrtf(s);
            mx = fmaxf(mx, norms[f]);
        }
        const float scale = 0.99f / mx;                    // U = 0.99
        for (int f = 0; f < Cout; ++f) {
            float dot = 0.f;
            for (int i = 0; i < d; ++i) dot += Wt[f * d + i] * scale * av[i];
            float n2 = norms[f] * scale; n2 = n2 * n2;
            float p = n2;                                  // [n2, n2^2, n2^4, n2^8, n2^16]
            dot += p * av[d + 0]; p *= p;
            dot += p * av[d + 1]; p *= p;
            dot += p * av[d + 2]; p *= p;
            dot += p * av[d + 3]; p *= p;
            dot += p * av[d + 4];
            const int hi = (int)floorf((dot + bb) * 10.0f);
            fh[f] = hi & 1;
        }
    }
}

// ===========================================================================
// K3: conv1 + ALSH mask + ReLU -> h1 (written even when masked: zeros feed the
// layer-2 patch means correctly).  Grid: (ceil(H1R*W1C/256), BATCH*COUT1).
// ===========================================================================
__global__ __launch_bounds__(256)
void conv1_kernel(const float* __restrict__ x, const float* __restrict__ W1g,
                  const int* __restrict__ qh1, const int* __restrict__ fh1,
                  float* __restrict__ h1)
{
    const int bc = blockIdx.y;
    const int b  = bc / COUT1, c = bc % COUT1;

    __shared__ float wsm[27];
    if (threadIdx.x < 27) wsm[threadIdx.x] = W1g[c * 27 + threadIdx.x];
    __syncthreads();

    const int idx = blockIdx.x * 256 + threadIdx.x;
    if (idx >= H1R * W1C) return;
    const int r = idx / W1C, w = idx % W1C;

    float val = 0.f;
    if (qh1[b] == fh1[c]) {
        float s = 0.f;
#pragma unroll
        for (int cin = 0; cin < C_IN; ++cin)
#pragma unroll
            for (int ki = 0; ki < 3; ++ki) {
                const float* p = x + (((size_t)b * C_IN + cin) * H_IN + (r + ki)) * W_IN + w;
#pragma unroll
                for (int kj = 0; kj < 3; ++kj)
                    s = fmaf(p[kj], wsm[(cin * 3 + ki) * 3 + kj], s);
            }
        val = fmaxf(s, 0.f);
    }
    h1[(((size_t)b * COUT1 + c) * H1R + r) * W1C + w] = val;
}

// conv2 at one (local row rl, local col cc) from the LDS tile.
__device__ __forceinline__
float conv2_lds(const float* __restrict__ tile, const float w2r[45], int rl, int cc)
{
    float s = 0.f;
#pragma unroll
    for (int cin = 0; cin < COUT1; ++cin)
#pragma unroll
        for (int ki = 0; ki < 3; ++ki) {
            const float* p = tile + (cin * TROWS + rl + ki) * TCOLS + cc;
#pragma unroll
            for (int kj = 0; kj < 3; ++kj)
                s = fmaf(p[kj], w2r[(cin * 3 + ki) * 3 + kj], s);
        }
    return s;
}

// ===========================================================================
// K6: fused conv2 + mask + ReLU + (row @ Wl^T + bl) with WMMA.
// Block = 160 threads (5 waves); wave w owns output channel c=w.
// The 1000-wide K dimension is processed in 25 chunks of 40 columns; the h1
// input tile for a chunk (5cin x 18rows x 42cols, 15.1KB) is staged in LDS and
// shared by all 5 waves (40x reuse vs. L2).  Wl lives in LDS (40KB).
// A-fragments follow the ISA 16x4 fp32 layout; each h2 element is computed
// exactly once.  Next chunk is software-prefetched (global_prefetch_b8).
// Grid: (16 row-tiles, 32 batches).
// ===========================================================================
__global__ __launch_bounds__(160)
void conv2_linear_kernel(const float* __restrict__ h1,
                         const float* __restrict__ W2g,
                         const float* __restrict__ Wl,
                         const float* __restrict__ bl,
                         const int* __restrict__ qh2,
                         const int* __restrict__ fh2,
                         float* __restrict__ out)
{
    __shared__ float wl_s[NLIN * W2C];                 // 40,000 B, Wl row-major [n][k]
    __shared__ float tile[TELEMS];                     // 15,120 B, [cin][rr][cc]

    for (int i = threadIdx.x; i < NLIN * W2C; i += 160) wl_s[i] = Wl[i];

    const int b    = blockIdx.y;
    const int r0   = blockIdx.x * 16;
    const int wv   = threadIdx.x >> 5;                 // wave id == output channel
    const int lane = threadIdx.x & 31;
    const int c    = wv;
    const int hh   = lane >> 4;                        // lane half (0/1)
    const int l    = lane & 15;

    const float* h1b = h1 + (size_t)b * COUT1 * H1R * W1C;

    float w2r[45];
#pragma unroll
    for (int i = 0; i < 45; ++i) w2r[i] = W2g[c * 45 + i];

    const bool active = (qh2[b] == fh2[c]);            // wave-uniform
    // Branch-free B-fragment: always-valid clamped LDS address + select.
    const int   nB   = (l < NLIN) ? l : 0;
    const float selB = (l < NLIN) ? 1.f : 0.f;

    v8f acc = {};

    for (int chunk = 0; chunk < NCHUNK; ++chunk) {
        const int base = chunk * WCHUNK;               // h1 col offset (<= 960)
        __syncthreads();                               // prev chunk consumers done
        // Cooperative fill: h1[b, cin, r0+rr, base+cc] -> tile, cc in [0,42).
        for (int i = threadIdx.x; i < TELEMS; i += 160) {
            const int cin = i / (TROWS * TCOLS);
            const int rem = i - cin * (TROWS * TCOLS);
            const int rr  = rem / TCOLS;
            const int cc  = rem - rr * TCOLS;
            const float* g = h1b + ((size_t)(cin * H1R + r0 + rr)) * W1C + (base + cc);
            tile[i] = *g;
            // Prefetch next chunk, one request per 128B cacheline.
            if (((i & 31) == 0) && (chunk + 1 < NCHUNK))
                __builtin_prefetch(g + WCHUNK, 0, 3);
        }
        __syncthreads();                               // tile (and wl_s) visible

        if (active) {
            // V_WMMA_F32_16X16X4_F32.
            // A 16x4 f32 layout: lane l (+16*hh) holds A[M=l][K=2*hh] (x) and
            // A[M=l][K=2*hh+1] (y).  B 4x16: mirrored K split, N = l.
            for (int kloc = 0; kloc < WCHUNK; kloc += 4) {
                const int cloc = kloc + 2 * hh;        // local col of this lane's pair
                v2f a;
                a.x = fmaxf(conv2_lds(tile, w2r, l, cloc),     0.f);
                a.y = fmaxf(conv2_lds(tile, w2r, l, cloc + 1), 0.f);
                const int kk = base + cloc;            // global K index
                v2f bf;
                bf.x = wl_s[nB * W2C + kk]     * selB;
                bf.y = wl_s[nB * W2C + kk + 1] * selB;
                acc = __builtin_amdgcn_wmma_f32_16x16x4_f32(
                    false, a, false, bf, (short)0, acc, false, false);
            }
        }
    }

    // C/D 16x16 f32 layout: VGPR v -> M = v + 8*hh, N = l.  Add bias; masked
    // channels (acc==0) still get +bl, matching  relu(y*mask) @ Wl^T + bl.
    const int n = l;
#pragma unroll
    for (int v = 0; v < 8; ++v) {
        const int m = v + (hh ? 8 : 0);
        if (n < NLIN)
            out[(((size_t)b * COUT2 + c) * H2R + (r0 + m)) * NLIN + n] = acc[v] + bl[n];
    }
}

// ===========================================================================
// Launcher
// ===========================================================================
extern "C" void kernel_launch(void* const* d_in, const int* in_sizes, int n_in,
                              void* d_out, int out_size, void* d_ws, size_t ws_size,
                              hipStream_t stream)
{
    (void)in_sizes; (void)n_in; (void)out_size; (void)ws_size;

    const float* x   = (const float*)d_in[0];
    const float* W1  = (const float*)d_in[1];
    const float* W2  = (const float*)d_in[2];
    const float* a1  = (const float*)d_in[3];
    const float* b1  = (const float*)d_in[4];
    const float* a2  = (const float*)d_in[5];
    const float* b2  = (const float*)d_in[6];
    const float* Wl  = (const float*)d_in[7];
    const float* bl  = (const float*)d_in[8];
    float*       out = (float*)d_out;

    // Workspace layout (all 4-byte aligned)
    float* h1  = (float*)d_ws;                 // 41,362,560 floats (~165 MB)
    float* q1s = h1  + H1_ELEMS;               // 32*27 patch-window sums
    float* q2s = q1s + BATCH * C_IN * 9;       // 32*45
    int*   qh1 = (int*)(q2s + BATCH * COUT1 * 9);
    int*   fh1 = qh1 + BATCH;
    int*   qh2 = fh1 + 8;
    int*   fh2 = qh2 + BATCH;

    // --- layer 1 hashes ---
    patchsum_kernel<<<BATCH * C_IN, 256, 0, stream>>>(
        x, C_IN, H_IN, W_IN, H1R, W1C, q1s);
    hash_kernel<<<1, 64, 0, stream>>>(
        q1s, C_IN, 1.0f / ((float)H1R * (float)W1C), W1, COUT1, a1, b1, qh1, fh1);

    // --- conv1 + mask + relu -> h1 ---
    {
        dim3 grid((H1R * W1C + 255) / 256, BATCH * COUT1);
        conv1_kernel<<<grid, 256, 0, stream>>>(x, W1, qh1, fh1, h1);
    }

    // --- layer 2 hashes (patch means of h1) ---
    patchsum_kernel<<<BATCH * COUT1, 256, 0, stream>>>(
        h1, COUT1, H1R, W1C, H2R, W2C, q2s);
    hash_kernel<<<1, 64, 0, stream>>>(
        q2s, COUT1, 1.0f / ((float)H2R * (float)W2C), W2, COUT2, a2, b2, qh2, fh2);

    // --- fused conv2 + mask + relu + linear (WMMA) -> out ---
    {
        dim3 grid(H2R / 16, BATCH);            // 16 row-tiles x 32 batches
        conv2_linear_kernel<<<grid, 160, 0, stream>>>(
            h1, W2, Wl, bl, qh2, fh2, out);
    }
}